// RelationAggregator_53206054863625
// MI455X (gfx1250) — compile-verified
//
#include <hip/hip_runtime.h>
#include <hip/hip_bf16.h>

// ---------------------------------------------------------------------------
// Relational GNN aggregation, algebraically restructured:
//   agg_r = segsum(w * x[src]) @ Wr + segsum(w) * br     (linearity hoist)
// Edge phase = pure scatter (bandwidth, L2-resident on MI455X's 192MB L2).
// GEMM phase = f32 WMMA (V_WMMA_F32_16X16X4_F32), wave32, 16x16 tiles.
// ---------------------------------------------------------------------------

#define N_NODES 50000
#define N_REL   4
#define N_EDGES 800000
#define DFEAT   128

typedef __attribute__((ext_vector_type(2))) float v2f;
typedef __attribute__((ext_vector_type(8))) float v8f;

#if defined(__has_builtin)
#  if __has_builtin(__builtin_amdgcn_wmma_f32_16x16x4_f32)
#    define HAS_WMMA_F32F32 1
#  endif
#endif
#ifndef HAS_WMMA_F32F32
#  define HAS_WMMA_F32F32 0
#endif

// ---------------------------------------------------------------------------
// Zero a float buffer (grid-stride).
// ---------------------------------------------------------------------------
__global__ __launch_bounds__(256) void zero_kernel(float* __restrict__ p, long n) {
    long i = (long)blockIdx.x * blockDim.x + threadIdx.x;
    long stride = (long)gridDim.x * blockDim.x;
    for (; i < n; i += stride) p[i] = 0.0f;
}

// ---------------------------------------------------------------------------
// rel_w = softmax(X @ Wrw + brw)   (N x 4, K=128) — one thread per node.
// ---------------------------------------------------------------------------
__global__ __launch_bounds__(256) void relw_kernel(const float* __restrict__ feat,
                                                   const float* __restrict__ Wrw,
                                                   const float* __restrict__ brw,
                                                   float* __restrict__ relw) {
    int n = blockIdx.x * blockDim.x + threadIdx.x;
    if (n >= N_NODES) return;
    float acc0 = brw[0], acc1 = brw[1], acc2 = brw[2], acc3 = brw[3];
    const float* f = feat + (size_t)n * DFEAT;
    #pragma unroll 4
    for (int k = 0; k < DFEAT; ++k) {
        float x = f[k];
        acc0 = fmaf(x, Wrw[k * 4 + 0], acc0);
        acc1 = fmaf(x, Wrw[k * 4 + 1], acc1);
        acc2 = fmaf(x, Wrw[k * 4 + 2], acc2);
        acc3 = fmaf(x, Wrw[k * 4 + 3], acc3);
    }
    float m = fmaxf(fmaxf(acc0, acc1), fmaxf(acc2, acc3));
    float e0 = __expf(acc0 - m), e1 = __expf(acc1 - m);
    float e2 = __expf(acc2 - m), e3 = __expf(acc3 - m);
    float inv = 1.0f / (e0 + e1 + e2 + e3);
    float* o = relw + (size_t)n * 4;
    o[0] = e0 * inv; o[1] = e1 * inv; o[2] = e2 * inv; o[3] = e3 * inv;
}

// ---------------------------------------------------------------------------
// Edge scatter for one relation:
//   H[dst, :] += w * feat[src, :]   (one wave of 32 lanes per edge, float4/lane)
//   sumw[dst] += w
// ---------------------------------------------------------------------------
__global__ __launch_bounds__(256) void scatter_kernel(const float* __restrict__ feat,
                                                      const long long* __restrict__ src_idx,
                                                      const long long* __restrict__ dst_idx,
                                                      const float* __restrict__ ew,
                                                      float* __restrict__ H,
                                                      float* __restrict__ sumw) {
    int lane = threadIdx.x & 31;
    int eid = blockIdx.x * (blockDim.x >> 5) + (threadIdx.x >> 5);
    if (eid >= N_EDGES) return;
    long long s = src_idx[eid];
    long long d = dst_idx[eid];
    float w = ew[eid];
    const float4 v = ((const float4*)(feat + (size_t)s * DFEAT))[lane];
    float* hd = H + (size_t)d * DFEAT + lane * 4;
    atomicAdd(hd + 0, w * v.x);
    atomicAdd(hd + 1, w * v.y);
    atomicAdd(hd + 2, w * v.z);
    atomicAdd(hd + 3, w * v.w);
    if (lane == 0) atomicAdd(sumw + d, w);
}

// ---------------------------------------------------------------------------
// f32 WMMA helper: one wave computes a 16x16 tile of A(16x128) @ B(128x128)
// using 32x V_WMMA_F32_16X16X4_F32.
//   A layout (16x4 f32): lanes 0-15 hold K=k,k+1 ; lanes 16-31 hold K=k+2,k+3
//   B layout (4x16 f32): mirrored, N = lane%16
//   C layout: VGPR v -> M = v (lanes 0-15) / v+8 (lanes 16-31), N = lane%16
// ---------------------------------------------------------------------------
__device__ __forceinline__ v8f wmma_tile_f32(const float* __restrict__ A, // row-major, ld=128
                                             const float* __restrict__ B, // row-major, ld=128
                                             int m0, int n0, int lane) {
    const int half = lane >> 4;   // 0: lanes 0-15, 1: lanes 16-31
    const int l16  = lane & 15;
    v8f c = {};
#if HAS_WMMA_F32F32
    #pragma unroll 8
    for (int k = 0; k < DFEAT; k += 4) {
        const int ka = k + half * 2;
        const float* arow = A + (size_t)(m0 + l16) * DFEAT + ka;
        v2f a; a.x = arow[0]; a.y = arow[1];
        const int kb = k + half * 2;
        v2f b;
        b.x = B[(size_t)kb * DFEAT + n0 + l16];
        b.y = B[(size_t)(kb + 1) * DFEAT + n0 + l16];
        c = __builtin_amdgcn_wmma_f32_16x16x4_f32(false, a, false, b,
                                                  (short)0, c, false, false);
    }
#else
    // Scalar fallback matching the C/D VGPR layout.
    #pragma unroll
    for (int v = 0; v < 8; ++v) {
        const int m = m0 + half * 8 + v;
        float acc = 0.0f;
        for (int k = 0; k < DFEAT; ++k)
            acc = fmaf(A[(size_t)m * DFEAT + k], B[(size_t)k * DFEAT + n0 + l16], acc);
        c[v] = acc;
    }
#endif
    return c;
}

// ---------------------------------------------------------------------------
// Per-relation GEMM + epilogue:
//   combined[m,n] += rel_w[m,r] * ( (H @ Wr)[m,n] + sumw[m]*br[n] )
// Grid: (N_NODES/16, 128/16), block = 32 (one wave per 16x16 tile).
// ---------------------------------------------------------------------------
__global__ __launch_bounds__(32) void rel_gemm_kernel(const float* __restrict__ H,
                                                      const float* __restrict__ W,   // 128x128
                                                      const float* __restrict__ br,  // 128
                                                      const float* __restrict__ sumw,
                                                      const float* __restrict__ relw,
                                                      int r,
                                                      float* __restrict__ combined) {
    const int m0 = blockIdx.x * 16;
    const int n0 = blockIdx.y * 16;
    const int lane = threadIdx.x;
    const int half = lane >> 4;
    const int l16  = lane & 15;

    v8f c = wmma_tile_f32(H, W, m0, n0, lane);

    const float brn = br[n0 + l16];
    #pragma unroll
    for (int v = 0; v < 8; ++v) {
        const int m = m0 + half * 8 + v;
        const float val = c[v] + sumw[m] * brn;
        combined[(size_t)m * DFEAT + n0 + l16] += relw[(size_t)m * 4 + r] * val;
    }
}

// ---------------------------------------------------------------------------
// Gate + output:
//   out[m,n] = sigmoid( (combined @ Wg)[m,n] + bg[n] ) * combined[m,n]
// ---------------------------------------------------------------------------
__global__ __launch_bounds__(32) void gate_kernel(const float* __restrict__ combined,
                                                  const float* __restrict__ Wg,  // 128x128
                                                  const float* __restrict__ bg,  // 128
                                                  float* __restrict__ out) {
    const int m0 = blockIdx.x * 16;
    const int n0 = blockIdx.y * 16;
    const int lane = threadIdx.x;
    const int half = lane >> 4;
    const int l16  = lane & 15;

    v8f c = wmma_tile_f32(combined, Wg, m0, n0, lane);

    const float bgn = bg[n0 + l16];
    #pragma unroll
    for (int v = 0; v < 8; ++v) {
        const int m = m0 + half * 8 + v;
        const float g = c[v] + bgn;
        const float s = 1.0f / (1.0f + __expf(-g));
        out[(size_t)m * DFEAT + n0 + l16] = s * combined[(size_t)m * DFEAT + n0 + l16];
    }
}

// ---------------------------------------------------------------------------
// Host-side orchestration. All launches on `stream`; graph-capture safe.
// ---------------------------------------------------------------------------
extern "C" void kernel_launch(void* const* d_in, const int* in_sizes, int n_in,
                              void* d_out, int out_size, void* d_ws, size_t ws_size,
                              hipStream_t stream) {
    (void)in_sizes; (void)n_in; (void)out_size; (void)ws_size;

    const float*      feat = (const float*)d_in[0];        // [N, 128]
    const long long*  eidx = (const long long*)d_in[1];    // [R, 2, E] int64
    const float*      ew   = (const float*)d_in[2];        // [R, E]
    const float*      Wr   = (const float*)d_in[3];        // [R, 128, 128]
    const float*      br   = (const float*)d_in[4];        // [R, 128]
    const float*      Wrw  = (const float*)d_in[5];        // [128, 4]
    const float*      brw  = (const float*)d_in[6];        // [4]
    const float*      Wg   = (const float*)d_in[7];        // [128, 128]
    const float*      bg   = (const float*)d_in[8];        // [128]
    float*            out  = (float*)d_out;                // [N, 128]

    // Workspace layout (floats): relw[N*4] | sumw[N] | H[N*128] | combined[N*128]
    float* relw     = (float*)d_ws;
    float* sumw     = relw + (size_t)N_NODES * 4;
    float* H        = sumw + (size_t)N_NODES;
    float* combined = H + (size_t)N_NODES * DFEAT;

    const int ZB = 2048; // zero-kernel blocks

    // combined = 0
    zero_kernel<<<ZB, 256, 0, stream>>>(combined, (long)N_NODES * DFEAT);

    // rel_w = softmax(X @ Wrw + brw)
    relw_kernel<<<(N_NODES + 255) / 256, 256, 0, stream>>>(feat, Wrw, brw, relw);

    dim3 gemm_grid(N_NODES / 16, DFEAT / 16);

    for (int r = 0; r < N_REL; ++r) {
        // H = 0, sumw = 0 (contiguous: sumw then H)
        zero_kernel<<<ZB, 256, 0, stream>>>(sumw, (long)N_NODES * (DFEAT + 1));

        const long long* src = eidx + ((size_t)r * 2 + 0) * N_EDGES;
        const long long* dst = eidx + ((size_t)r * 2 + 1) * N_EDGES;

        scatter_kernel<<<(N_EDGES + 7) / 8, 256, 0, stream>>>(
            feat, src, dst, ew + (size_t)r * N_EDGES, H, sumw);

        rel_gemm_kernel<<<gemm_grid, 32, 0, stream>>>(
            H, Wr + (size_t)r * DFEAT * DFEAT, br + (size_t)r * DFEAT,
            sumw, relw, r, combined);
    }

    gate_kernel<<<gemm_grid, 32, 0, stream>>>(combined, Wg, bg, out);
}